// MolecularEncoder_44083544326762
// MI455X (gfx1250) — compile-verified
//
#include <hip/hip_runtime.h>
#include <hip/hip_bf16.h>

#define N_NODES   250000
#define N_EDGES   500000
#define N_GRAPHS  8192
#define HIDDEN    128
#define PROJ_HID  256
#define NUM_LAYERS 5
#define BN_EPS    1e-5f

typedef __attribute__((ext_vector_type(2))) float v2f;
typedef __attribute__((ext_vector_type(8))) float v8f;

struct AtomTabs { const float* t[9]; };
struct BondTabs { const float* t[4]; };

// D = A(16x4) * B(4x16) + C, fp32 WMMA (native CDNA5 fp32 matrix op)
__device__ __forceinline__ v8f wmma4(v2f a, v2f b, v8f c) {
    return __builtin_amdgcn_wmma_f32_16x16x4_f32(
        /*neg_a=*/false, a, /*neg_b=*/false, b,
        /*c_mod=*/(short)0, c, /*reuse_a=*/false, /*reuse_b=*/false);
}

// ---------------------------------------------------------------- zero fill
__global__ void k_zero4(float4* __restrict__ p, long n4) {
    long i = (long)blockIdx.x * blockDim.x + threadIdx.x;
    if (i < n4) p[i] = make_float4(0.f, 0.f, 0.f, 0.f);
}

// ------------------------------------------------- atom encoder: h = sum emb
__global__ void k_atom_encode(const int* __restrict__ x, AtomTabs tabs,
                              float* __restrict__ h) {
    long gid = (long)blockIdx.x * blockDim.x + threadIdx.x; // N*32 threads
    if (gid >= (long)N_NODES * 32) return;
    int node = (int)(gid >> 5);
    int c    = (int)(gid & 31) << 2;
    const int* xr = x + (size_t)node * 9;
    float4 acc = make_float4(0.f, 0.f, 0.f, 0.f);
#pragma unroll
    for (int i = 0; i < 9; ++i) {
        float4 v = *(const float4*)(tabs.t[i] + (size_t)xr[i] * HIDDEN + c);
        acc.x += v.x; acc.y += v.y; acc.z += v.z; acc.w += v.w;
    }
    *(float4*)(h + (size_t)node * HIDDEN + c) = acc;
}

// --------------------- message + scatter: agg[dst] += relu(h[src] + bondemb)
__global__ void k_message(const float* __restrict__ h,
                          const int* __restrict__ edge_index,
                          const int* __restrict__ edge_attr,
                          BondTabs tabs, float* __restrict__ agg) {
    long gid = (long)blockIdx.x * blockDim.x + threadIdx.x; // E*32 threads
    if (gid >= (long)N_EDGES * 32) return;
    int e = (int)(gid >> 5);
    int c = (int)(gid & 31) << 2;
    int src = edge_index[e];
    int dst = edge_index[N_EDGES + e];
    const int* ea = edge_attr + (size_t)e * 4;
    float4 ev = make_float4(0.f, 0.f, 0.f, 0.f);
#pragma unroll
    for (int i = 0; i < 4; ++i) { // tiny tables: ~7.7KB total, L0/L2 resident
        float4 v = *(const float4*)(tabs.t[i] + (size_t)ea[i] * HIDDEN + c);
        ev.x += v.x; ev.y += v.y; ev.z += v.z; ev.w += v.w;
    }
    float4 hv = *(const float4*)(h + (size_t)src * HIDDEN + c); // L2-resident gather
    float m0 = fmaxf(hv.x + ev.x, 0.f);
    float m1 = fmaxf(hv.y + ev.y, 0.f);
    float m2 = fmaxf(hv.z + ev.z, 0.f);
    float m3 = fmaxf(hv.w + ev.w, 0.f);
    float* a = agg + (size_t)dst * HIDDEN + c;
    atomicAdd(a + 0, m0);
    atomicAdd(a + 1, m1);
    atomicAdd(a + 2, m2);
    atomicAdd(a + 3, m3);
}

// ---- GINE MLP: z = relu(((1+eps)h + agg) @ W1 + b1) @ W2 + b2  (16 nodes/blk)
#define LDA 132  // 128 + 4 pad: column reads hit distinct LDS banks
__global__ __launch_bounds__(256)
void k_gine_mlp(const float* __restrict__ h, const float* __restrict__ agg,
                const float* __restrict__ epsp,
                const float* __restrict__ W1, const float* __restrict__ b1,
                const float* __restrict__ W2, const float* __restrict__ b2,
                float* __restrict__ z) {
    __shared__ float zin[16 * LDA];
    __shared__ float tmid[16 * LDA];
    const int tid   = threadIdx.x;
    const int node0 = blockIdx.x * 16;          // 250000/16 = 15625 exact
    const float scale = 1.0f + epsp[0];

    for (int idx = tid; idx < 16 * 128; idx += 256) {
        int r = idx >> 7, c = idx & 127;
        size_t off = (size_t)(node0 + r) * HIDDEN + c;
        zin[r * LDA + c] = scale * h[off] + agg[off];
    }
    __syncthreads();

    const int lane = tid & 31;
    const int wave = tid >> 5;                  // 8 waves -> 8 col tiles of 16
    const int m    = lane & 15;                 // A row / B col / D col
    const int kh   = (lane >> 4) << 1;          // K sub-pair select
    const int colt = wave << 4;
    const int rb   = (lane >> 4) << 3;          // D row base

    { // GEMM1: (16x128) @ (128x128) -> relu tile in LDS
        v8f acc = {};
#pragma unroll
        for (int k = 0; k < 128; k += 4) {
            v2f a = { zin[m * LDA + k + kh], zin[m * LDA + k + kh + 1] };
            v2f b = { W1[(size_t)(k + kh) * 128 + colt + m],
                      W1[(size_t)(k + kh + 1) * 128 + colt + m] };
            acc = wmma4(a, b, acc);
        }
        float bias = b1[colt + m];
#pragma unroll
        for (int v = 0; v < 8; ++v)
            tmid[(rb + v) * LDA + colt + m] = fmaxf(acc[v] + bias, 0.f);
    }
    __syncthreads();

    { // GEMM2: (16x128) @ (128x128) -> z
        v8f acc = {};
#pragma unroll
        for (int k = 0; k < 128; k += 4) {
            v2f a = { tmid[m * LDA + k + kh], tmid[m * LDA + k + kh + 1] };
            v2f b = { W2[(size_t)(k + kh) * 128 + colt + m],
                      W2[(size_t)(k + kh + 1) * 128 + colt + m] };
            acc = wmma4(a, b, acc);
        }
        float bias = b2[colt + m];
#pragma unroll
        for (int v = 0; v < 8; ++v)
            z[(size_t)(node0 + rb + v) * HIDDEN + colt + m] = acc[v] + bias;
    }
}

// ------------------------------------------------ BatchNorm statistics pass
__global__ void k_bn_stats(const float* __restrict__ z, float* __restrict__ stats) {
    __shared__ float ssum[256], ssq[256];
    int tid  = threadIdx.x;
    int ch   = tid & 127;
    int slot = tid >> 7;
    float s = 0.f, q = 0.f;
    for (int node = blockIdx.x * 2 + slot; node < N_NODES; node += gridDim.x * 2) {
        float v = z[(size_t)node * HIDDEN + ch];
        s += v; q += v * v;
    }
    ssum[tid] = s; ssq[tid] = q;
    __syncthreads();
    if (slot == 0) {
        atomicAdd(&stats[ch],       ssum[tid] + ssum[tid + 128]);
        atomicAdd(&stats[128 + ch], ssq[tid]  + ssq[tid + 128]);
    }
}

__global__ void k_bn_finalize(float* __restrict__ stats) {
    int ch = threadIdx.x;
    if (ch >= 128) return;
    const float invN = 1.0f / (float)N_NODES;
    float mu  = stats[ch] * invN;
    float var = stats[128 + ch] * invN - mu * mu;
    stats[256 + ch] = mu;
    stats[384 + ch] = rsqrtf(var + BN_EPS);
}

// ------------------------------------- h = relu(gamma*(z-mu)*rstd + beta)
__global__ void k_bn_apply(const float* __restrict__ z, const float* __restrict__ stats,
                           const float* __restrict__ gamma, const float* __restrict__ beta,
                           float* __restrict__ h) {
    long gid = (long)blockIdx.x * blockDim.x + threadIdx.x;
    if (gid >= (long)N_NODES * 32) return;
    int node = (int)(gid >> 5);
    int c    = (int)(gid & 31) << 2;
    float4 zv = *(const float4*)(z + (size_t)node * HIDDEN + c);
    float4 mu = *(const float4*)(stats + 256 + c);
    float4 iv = *(const float4*)(stats + 384 + c);
    float4 g  = *(const float4*)(gamma + c);
    float4 b  = *(const float4*)(beta + c);
    float4 o;
    o.x = fmaxf(g.x * (zv.x - mu.x) * iv.x + b.x, 0.f);
    o.y = fmaxf(g.y * (zv.y - mu.y) * iv.y + b.y, 0.f);
    o.z = fmaxf(g.z * (zv.z - mu.z) * iv.z + b.z, 0.f);
    o.w = fmaxf(g.w * (zv.w - mu.w) * iv.w + b.w, 0.f);
    *(float4*)(h + (size_t)node * HIDDEN + c) = o;
}

// ------------------------------------------------------- graph mean pooling
__global__ void k_pool_accum(const float* __restrict__ h, const int* __restrict__ batch,
                             float* __restrict__ pooled, float* __restrict__ counts) {
    long gid = (long)blockIdx.x * blockDim.x + threadIdx.x;
    if (gid >= (long)N_NODES * 32) return;
    int node = (int)(gid >> 5);
    int c    = (int)(gid & 31) << 2;
    int g    = batch[node];
    float4 hv = *(const float4*)(h + (size_t)node * HIDDEN + c);
    float* p = pooled + (size_t)g * HIDDEN + c;
    atomicAdd(p + 0, hv.x);
    atomicAdd(p + 1, hv.y);
    atomicAdd(p + 2, hv.z);
    atomicAdd(p + 3, hv.w);
    if ((gid & 31) == 0) atomicAdd(&counts[g], 1.0f);
}

__global__ void k_pool_div(float* __restrict__ pooled, const float* __restrict__ counts) {
    long gid = (long)blockIdx.x * blockDim.x + threadIdx.x;
    if (gid >= (long)N_GRAPHS * 32) return;
    int g = (int)(gid >> 5);
    int c = (int)(gid & 31) << 2;
    float inv = 1.0f / fmaxf(counts[g], 1.0f);
    float4 v = *(float4*)(pooled + (size_t)g * HIDDEN + c);
    v.x *= inv; v.y *= inv; v.z *= inv; v.w *= inv;
    *(float4*)(pooled + (size_t)g * HIDDEN + c) = v;
}

// ------------------- projection head: relu(pooled@W1+b1)@W2+b2, 16 graphs/blk
#define LDT 260  // 256 + 4 pad
__global__ __launch_bounds__(256)
void k_proj(const float* __restrict__ pooled,
            const float* __restrict__ W1, const float* __restrict__ b1,
            const float* __restrict__ W2, const float* __restrict__ b2,
            float* __restrict__ out) {
    __shared__ float ain[16 * LDA];
    __shared__ float tmid[16 * LDT];
    const int tid = threadIdx.x;
    const int g0  = blockIdx.x * 16;            // 8192/16 = 512 exact

    for (int idx = tid; idx < 16 * 128; idx += 256) {
        int r = idx >> 7, c = idx & 127;
        ain[r * LDA + c] = pooled[(size_t)(g0 + r) * HIDDEN + c];
    }
    __syncthreads();

    const int lane = tid & 31;
    const int wave = tid >> 5;
    const int m    = lane & 15;
    const int kh   = (lane >> 4) << 1;
    const int rb   = (lane >> 4) << 3;

    // GEMM1: (16x128)@(128x256); 16 col tiles across 8 waves -> 2 tiles each
#pragma unroll
    for (int tt = 0; tt < 2; ++tt) {
        int colt = (wave + tt * 8) << 4;
        v8f acc = {};
#pragma unroll
        for (int k = 0; k < 128; k += 4) {
            v2f a = { ain[m * LDA + k + kh], ain[m * LDA + k + kh + 1] };
            v2f b = { W1[(size_t)(k + kh) * PROJ_HID + colt + m],
                      W1[(size_t)(k + kh + 1) * PROJ_HID + colt + m] };
            acc = wmma4(a, b, acc);
        }
        float bias = b1[colt + m];
#pragma unroll
        for (int v = 0; v < 8; ++v)
            tmid[(rb + v) * LDT + colt + m] = fmaxf(acc[v] + bias, 0.f);
    }
    __syncthreads();

    { // GEMM2: (16x256)@(256x128), K=256
        int colt = wave << 4;
        v8f acc = {};
#pragma unroll
        for (int k = 0; k < 256; k += 4) {
            v2f a = { tmid[m * LDT + k + kh], tmid[m * LDT + k + kh + 1] };
            v2f b = { W2[(size_t)(k + kh) * HIDDEN + colt + m],
                      W2[(size_t)(k + kh + 1) * HIDDEN + colt + m] };
            acc = wmma4(a, b, acc);
        }
        float bias = b2[colt + m];
#pragma unroll
        for (int v = 0; v < 8; ++v)
            out[(size_t)(g0 + rb + v) * HIDDEN + colt + m] = acc[v] + bias;
    }
}

// ----------------------------------------------------- row-wise L2 normalize
__global__ void k_l2norm(float* __restrict__ out) {
    __shared__ float red[128];
    int g = blockIdx.x;
    int c = threadIdx.x;
    float v = out[(size_t)g * HIDDEN + c];
    red[c] = v * v;
    __syncthreads();
    for (int s = 64; s > 0; s >>= 1) {
        if (c < s) red[c] += red[c + s];
        __syncthreads();
    }
    float norm = fmaxf(sqrtf(red[0]), 1e-8f);
    out[(size_t)g * HIDDEN + c] = v / norm;
}

// ===========================================================================
extern "C" void kernel_launch(void* const* d_in, const int* in_sizes, int n_in,
                              void* d_out, int out_size, void* d_ws, size_t ws_size,
                              hipStream_t stream) {
    (void)in_sizes; (void)n_in; (void)out_size; (void)ws_size;
    int i = 0;
    const int* x          = (const int*)d_in[i++]; // [N,9]
    const int* edge_index = (const int*)d_in[i++]; // [2,E]
    const int* batch      = (const int*)d_in[i++]; // [N]
    const int* edge_attr  = (const int*)d_in[i++]; // [E,4]
    AtomTabs at; for (int k = 0; k < 9; ++k) at.t[k] = (const float*)d_in[i++];
    BondTabs bt; for (int k = 0; k < 4; ++k) bt.t[k] = (const float*)d_in[i++];
    struct Layer { const float *eps, *W1, *b1, *W2, *b2, *gamma, *beta; } L[NUM_LAYERS];
    for (int l = 0; l < NUM_LAYERS; ++l) {
        L[l].eps   = (const float*)d_in[i++];
        L[l].W1    = (const float*)d_in[i++];
        L[l].b1    = (const float*)d_in[i++];
        L[l].W2    = (const float*)d_in[i++];
        L[l].b2    = (const float*)d_in[i++];
        L[l].gamma = (const float*)d_in[i++];
        L[l].beta  = (const float*)d_in[i++];
    }
    const float* PW1 = (const float*)d_in[i++];
    const float* pb1 = (const float*)d_in[i++];
    const float* PW2 = (const float*)d_in[i++];
    const float* pb2 = (const float*)d_in[i++];

    // workspace partition (fp32): h, agg, z (each N*128) + stats + pooled + counts
    float* ws     = (float*)d_ws;
    float* h      = ws;
    float* agg    = h   + (size_t)N_NODES * HIDDEN;
    float* z      = agg + (size_t)N_NODES * HIDDEN;
    float* stats  = z   + (size_t)N_NODES * HIDDEN;   // 512 floats
    float* pooled = stats + 512;                      // [G,128]
    float* counts = pooled + (size_t)N_GRAPHS * HIDDEN;

    const long nNH32 = (long)N_NODES * 32;            // (node, 4-ch) work items
    const long nEH32 = (long)N_EDGES * 32;
    dim3 blk(256);

    // h = atom encoder
    k_atom_encode<<<dim3((nNH32 + 255) / 256), blk, 0, stream>>>(x, at, h);

    for (int l = 0; l < NUM_LAYERS; ++l) {
        long agg4 = (long)N_NODES * HIDDEN / 4;
        k_zero4<<<dim3((agg4 + 255) / 256), blk, 0, stream>>>((float4*)agg, agg4);
        k_message<<<dim3((nEH32 + 255) / 256), blk, 0, stream>>>(
            h, edge_index, edge_attr, bt, agg);
        k_gine_mlp<<<dim3(N_NODES / 16), blk, 0, stream>>>(
            h, agg, L[l].eps, L[l].W1, L[l].b1, L[l].W2, L[l].b2, z);
        k_zero4<<<dim3(1), dim3(128), 0, stream>>>((float4*)stats, 128);
        k_bn_stats<<<dim3(512), blk, 0, stream>>>(z, stats);
        k_bn_finalize<<<dim3(1), dim3(128), 0, stream>>>(stats);
        k_bn_apply<<<dim3((nNH32 + 255) / 256), blk, 0, stream>>>(
            z, stats, L[l].gamma, L[l].beta, h);
    }

    // mean pool per graph
    long pool4 = ((long)N_GRAPHS * HIDDEN + N_GRAPHS) / 4; // pooled + counts contiguous
    k_zero4<<<dim3((pool4 + 255) / 256), blk, 0, stream>>>((float4*)pooled, pool4);
    k_pool_accum<<<dim3((nNH32 + 255) / 256), blk, 0, stream>>>(h, batch, pooled, counts);
    k_pool_div<<<dim3(((long)N_GRAPHS * 32 + 255) / 256), blk, 0, stream>>>(pooled, counts);

    // projection + L2 normalize
    k_proj<<<dim3(N_GRAPHS / 16), blk, 0, stream>>>(pooled, PW1, pb1, PW2, pb2, (float*)d_out);
    k_l2norm<<<dim3(N_GRAPHS), dim3(128), 0, stream>>>((float*)d_out);
}